// MaxPooling2_7541962572231
// MI455X (gfx1250) — compile-verified
//
#include <hip/hip_runtime.h>
#include <stdint.h>

#define BLK 256
#define STILE 4096            // 256 threads * 16 items
#define CAPV (1u << 22)       // voxel-id keyspace cap (actual ~1e5)

static __device__ __forceinline__ unsigned fenc(float f) {
  unsigned u = __float_as_uint(f);
  return (u & 0x80000000u) ? ~u : (u | 0x80000000u);   // order-preserving
}
static __device__ __forceinline__ float fdec(unsigned u) {
  unsigned v = (u & 0x80000000u) ? (u & 0x7fffffffu) : ~u;
  return __uint_as_float(v);
}

// ---------------- generic fills ----------------
__global__ void k_fill_u32(unsigned* p, unsigned v, long long nE) {
  long long i = (long long)blockIdx.x * blockDim.x + threadIdx.x;
  long long st = (long long)gridDim.x * blockDim.x;
  for (; i < nE; i += st) p[i] = v;
}
__global__ void k_init_slots(unsigned* slots) {
  slots[0] = 0xFFFFFFFFu; slots[1] = 0xFFFFFFFFu;  // min enc
  slots[2] = 0u;          slots[3] = 0u;           // max enc
  slots[4] = 0u;          slots[5] = 0u;           // n_clusters, n_groups
}

// ---------------- bbox reduction ----------------
__global__ void k_minmax(const float* __restrict__ pos, int n, unsigned* slots) {
  __shared__ unsigned sm[4][BLK];
  unsigned mn0 = 0xFFFFFFFFu, mn1 = 0xFFFFFFFFu, mx0 = 0u, mx1 = 0u;
  long long i = (long long)blockIdx.x * blockDim.x + threadIdx.x;
  long long st = (long long)gridDim.x * blockDim.x;
  for (; i < n; i += st) {
    unsigned e0 = fenc(pos[i * 3 + 1]);
    unsigned e1 = fenc(pos[i * 3 + 2]);
    mn0 = min(mn0, e0); mx0 = max(mx0, e0);
    mn1 = min(mn1, e1); mx1 = max(mx1, e1);
  }
  int t = threadIdx.x;
  sm[0][t] = mn0; sm[1][t] = mn1; sm[2][t] = mx0; sm[3][t] = mx1;
  __syncthreads();
  for (int o = BLK / 2; o > 0; o >>= 1) {
    if (t < o) {
      sm[0][t] = min(sm[0][t], sm[0][t + o]);
      sm[1][t] = min(sm[1][t], sm[1][t + o]);
      sm[2][t] = max(sm[2][t], sm[2][t + o]);
      sm[3][t] = max(sm[3][t], sm[3][t + o]);
    }
    __syncthreads();
  }
  if (t == 0) {
    atomicMin(&slots[0], sm[0][0]);
    atomicMin(&slots[1], sm[1][0]);
    atomicMax(&slots[2], sm[2][0]);
    atomicMax(&slots[3], sm[3][0]);
  }
}

// ---------------- voxel id + histogram ----------------
__global__ void k_raw(const float* __restrict__ pos, const int* __restrict__ batch,
                      int n, const unsigned* slots, unsigned* raw, unsigned* counts) {
  long long i = (long long)blockIdx.x * blockDim.x + threadIdx.x;
  if (i >= n) return;
  float s0 = fdec(slots[0]), s1 = fdec(slots[1]);
  float e0 = fdec(slots[2]), e1 = fdec(slots[3]);
  int d0 = (int)(floorf((e0 - s0) / 5.0f) + 1.0f);
  int d1 = (int)(floorf((e1 - s1) / 5.0f) + 1.0f);
  int cx = (int)floorf((pos[i * 3 + 1] - s0) / 5.0f);
  int cy = (int)floorf((pos[i * 3 + 2] - s1) / 5.0f);
  long long r = (long long)batch[i] * (long long)(d0 * d1) + (long long)cx * d1 + cy;
  unsigned ru = (r < 0) ? 0u : (r >= (long long)CAPV ? (CAPV - 1u) : (unsigned)r);
  raw[i] = ru;
  atomicAdd(&counts[ru], 1u);
}

// ---------------- exclusive scan pipeline (u32) ----------------
__global__ void k_scan_blocks(const unsigned* __restrict__ in, unsigned* __restrict__ out,
                              unsigned* bsums, long long nLen, int mode) {
  __shared__ unsigned s[BLK];
  int t = threadIdx.x;
  long long base = (long long)blockIdx.x * STILE + (long long)t * 16;
  unsigned v[16]; unsigned tsum = 0;
#pragma unroll
  for (int k = 0; k < 16; k++) {
    long long i = base + k;
    unsigned u = (i < nLen) ? in[i] : 0u;
    if (mode) u = u ? 1u : 0u;
    v[k] = u; tsum += u;
  }
  s[t] = tsum; __syncthreads();
  for (int o = 1; o < BLK; o <<= 1) {
    unsigned u = (t >= o) ? s[t - o] : 0u;
    __syncthreads();
    s[t] += u;
    __syncthreads();
  }
  unsigned run = s[t] - tsum;
#pragma unroll
  for (int k = 0; k < 16; k++) {
    long long i = base + k;
    if (i < nLen) out[i] = run;
    run += v[k];
  }
  if (t == BLK - 1) bsums[blockIdx.x] = s[BLK - 1];
}
__global__ void k_scan_carry(unsigned* b, int m) {
  __shared__ unsigned s[BLK];
  __shared__ unsigned carry;
  int t = threadIdx.x;
  if (t == 0) carry = 0;
  __syncthreads();
  for (int base = 0; base < m; base += BLK) {
    int i = base + t;
    unsigned v = (i < m) ? b[i] : 0u;
    s[t] = v; __syncthreads();
    for (int o = 1; o < BLK; o <<= 1) {
      unsigned u = (t >= o) ? s[t - o] : 0u;
      __syncthreads();
      s[t] += u;
      __syncthreads();
    }
    unsigned excl = s[t] - v;
    unsigned tot = s[BLK - 1];
    if (i < m) b[i] = carry + excl;
    __syncthreads();
    if (t == 0) carry += tot;
    __syncthreads();
  }
}
__global__ void k_scan_add(unsigned* out, const unsigned* bsums, long long nLen) {
  unsigned add = bsums[blockIdx.x];
  long long base = (long long)blockIdx.x * STILE + (long long)threadIdx.x * 16;
#pragma unroll
  for (int k = 0; k < 16; k++) {
    long long i = base + k;
    if (i < nLen) out[i] += add;
  }
}

__global__ void k_total_p(const unsigned* P, const unsigned* counts, unsigned* slots) {
  slots[4] = P[CAPV - 1] + (counts[CAPV - 1] ? 1u : 0u);
}

// ---------------- cluster assignment + node accumulation ----------------
__global__ void k_cluster(const unsigned* __restrict__ raw, const unsigned* __restrict__ P,
                          unsigned* cluster, const float* __restrict__ pos,
                          const int* __restrict__ batch, unsigned* cnt, int* bmax,
                          unsigned long long* psum, int n) {
  long long i = (long long)blockIdx.x * blockDim.x + threadIdx.x;
  if (i >= n) return;
  unsigned c = P[raw[i]];
  cluster[i] = c;
  atomicAdd(&cnt[c], 1u);
  atomicMax(&bmax[c], batch[i]);
#pragma unroll
  for (int k = 0; k < 3; k++) {
    double v = (double)pos[i * 3 + k];
    long long q = (long long)llrint(v * 1048576.0);   // 2^20 fixed point
    atomicAdd(&psum[(size_t)c * 3 + k], (unsigned long long)q);
  }
}
__global__ void k_xmax(const float* __restrict__ x, const unsigned* __restrict__ cluster,
                       unsigned* acc, long long n64) {
  long long j = (long long)blockIdx.x * blockDim.x + threadIdx.x;
  if (j >= n64) return;
  unsigned c = cluster[j >> 6];
  atomicMax(&acc[((size_t)c << 6) + (j & 63)], fenc(x[j]));
}
__global__ void k_xdec(float* outx, const unsigned* slots, long long n64) {
  long long j = (long long)blockIdx.x * blockDim.x + threadIdx.x;
  if (j >= n64) return;
  unsigned ncl = slots[4];
  unsigned a = ((const unsigned*)outx)[j];
  outx[j] = ((unsigned)(j >> 6) < ncl) ? fdec(a) : 0.0f;
}
__global__ void k_node_out(const unsigned* slots, const unsigned* __restrict__ cnt,
                           const int* __restrict__ bmax, const unsigned long long* __restrict__ psum,
                           float* outPos, float* outB, int n) {
  long long c = (long long)blockIdx.x * blockDim.x + threadIdx.x;
  if (c >= n) return;
  unsigned ncl = slots[4];
  if ((unsigned)c < ncl) {
    float fc = (float)cnt[c]; if (fc < 1.0f) fc = 1.0f;
    float m0 = (float)((double)(long long)psum[c * 3 + 0] / 1048576.0) / fc;
    float m1 = (float)((double)(long long)psum[c * 3 + 1] / 1048576.0) / fc;
    float m2 = (float)((double)(long long)psum[c * 3 + 2] / 1048576.0) / fc;
    outPos[c * 3 + 0] = m0;
    outPos[c * 3 + 1] = floorf(m1 / 4.0f);
    outPos[c * 3 + 2] = floorf(m2 / 4.0f);
    outB[c] = (float)bmax[c];
  } else {
    outPos[c * 3 + 0] = 0.0f; outPos[c * 3 + 1] = 0.0f; outPos[c * 3 + 2] = 0.0f;
    outB[c] = -1.0f;
  }
}

// ---------------- edge keys ----------------
__global__ void k_edge_key(const int* __restrict__ ei, const unsigned* __restrict__ cluster,
                           int e, int n, unsigned long long* keyA, unsigned* payA) {
  long long g = (long long)blockIdx.x * blockDim.x + threadIdx.x;
  if (g >= e) return;
  unsigned cs = cluster[ei[g]];
  unsigned cd = cluster[ei[(long long)e + g]];
  if (cs == cd) { cs = (unsigned)n; cd = (unsigned)n; }     // self-loops -> sort last, invalid
  keyA[g] = ((unsigned long long)cs << 20) | (unsigned long long)cd;
  payA[g] = (unsigned)g;
}

// ---------------- radix sort: histogram ----------------
__global__ void k_rhist(const unsigned long long* __restrict__ key, long long nLen,
                        int shift, unsigned* hist, int numBlocks) {
  __shared__ unsigned h[256];
  int t = threadIdx.x;
  h[t] = 0;
  __syncthreads();
  long long base = (long long)blockIdx.x * STILE;
  for (int k = 0; k < 16; k++) {
    long long idx = base + (long long)k * BLK + t;
    if (idx < nLen) {
      unsigned d = (unsigned)((key[idx] >> shift) & 0xFFull);
      atomicAdd(&h[d], 1u);
    }
  }
  __syncthreads();
  hist[(long long)t * numBlocks + blockIdx.x] = h[t];  // bin-major
}

// ---------------- radix sort: stable scatter, async-LDS staged keys ----------------
__global__ void k_rscatter(const unsigned long long* __restrict__ keyIn,
                           const unsigned* __restrict__ payIn,
                           unsigned long long* __restrict__ keyOut,
                           unsigned* __restrict__ payOut,
                           long long nLen, int shift,
                           const unsigned* __restrict__ histS, int numBlocks) {
  __shared__ unsigned long long skey[STILE];   // 32KB
  __shared__ unsigned waveCnt[8 * 256];        // 8KB
  __shared__ unsigned running[256];            // 1KB
  int t = threadIdx.x;
  int wave = t >> 5, lane = t & 31;
  long long base = (long long)blockIdx.x * STILE;

  // CDNA5 async global->LDS staging of this block's key tile (drained before any rank math)
  unsigned long long gbase = (unsigned long long)(uintptr_t)keyIn;
  for (int k = 0; k < 16; k++) {
    long long idx = base + (long long)k * BLK + t;
    if (idx < nLen) {
      unsigned ldsoff = (unsigned)(uintptr_t)&skey[k * BLK + t];
      unsigned voff = (unsigned)(idx * 8);
      asm volatile("global_load_async_to_lds_b64 %0, %1, %2"
                   :: "v"(ldsoff), "v"(voff), "s"(gbase) : "memory");
    }
  }
  asm volatile("s_wait_asynccnt 0" ::: "memory");
  running[t] = 0;
  __syncthreads();

  unsigned ltmask = (1u << lane) - 1u;
  for (int s = 0; s < 16; s++) {
    for (int z = t; z < 8 * 256; z += BLK) waveCnt[z] = 0;
    __syncthreads();
    long long idx = base + (long long)s * BLK + t;
    bool active = idx < nLen;
    unsigned long long key = active ? skey[s * BLK + t] : 0ull;
    unsigned dig = (unsigned)((key >> shift) & 0xFFull);
    unsigned activeBal = (unsigned)__ballot(active);
    unsigned mask = activeBal;
#pragma unroll
    for (int b = 0; b < 8; b++) {
      unsigned bit = (dig >> b) & 1u;
      unsigned bal = (unsigned)__ballot(active && bit);
      mask &= bit ? bal : ~bal;
    }
    unsigned wrank = __popc(mask & ltmask);
    unsigned wcnt = __popc(mask);
    if (active && wrank == 0) waveCnt[wave * 256 + dig] = wcnt;
    __syncthreads();
    if (active) {
      unsigned cross = 0;
      for (int w = 0; w < wave; w++) cross += waveCnt[w * 256 + dig];
      unsigned rankTile = running[dig] + cross + wrank;
      unsigned pos = histS[(long long)dig * numBlocks + blockIdx.x] + rankTile;
      keyOut[pos] = key;
      payOut[pos] = payIn[idx];
    }
    __syncthreads();
    unsigned add = 0;
    for (int w = 0; w < 8; w++) add += waveCnt[w * 256 + t];
    running[t] += add;
    __syncthreads();
  }
}

// ---------------- group segmentation ----------------
__global__ void k_heads(const unsigned long long* __restrict__ key, unsigned* head, int e) {
  long long j = (long long)blockIdx.x * blockDim.x + threadIdx.x;
  if (j >= e) return;
  head[j] = (j == 0 || key[j] != key[j - 1]) ? 1u : 0u;
}
__global__ void k_ngroups(const unsigned* headA, const unsigned* gidEx,
                          unsigned* gstart, unsigned* slots, int e) {
  unsigned ng = gidEx[e - 1] + headA[e - 1];
  slots[5] = ng;
  gstart[ng] = (unsigned)e;
}
__global__ void k_gstart(const unsigned* __restrict__ headA, const unsigned* __restrict__ gidEx,
                         unsigned* gstart, int e) {
  long long j = (long long)blockIdx.x * blockDim.x + threadIdx.x;
  if (j >= e) return;
  if (headA[j]) gstart[gidEx[j]] = (unsigned)j;
}
__global__ void k_groups(const unsigned long long* __restrict__ keyS,
                         const unsigned* __restrict__ payS,
                         const unsigned* __restrict__ gstart, const unsigned* slots,
                         const float* __restrict__ ea, float* outEI, float* outEA,
                         int e, int n) {
  long long j = (long long)blockIdx.x * blockDim.x + threadIdx.x;
  if (j >= (long long)e * 8) return;
  long long g = j >> 3;
  int f = (int)(j & 7);
  unsigned ng = slots[5];
  if ((unsigned long long)g < ng) {
    unsigned j0 = gstart[g], j1 = gstart[g + 1];
    unsigned long long key = keyS[j0];
    unsigned cs = (unsigned)(key >> 20);
    unsigned cd = (unsigned)(key & 0xFFFFFull);
    bool valid = cs < (unsigned)n;
    float sum = 0.0f;
    for (unsigned q = j0; q < j1; q++) sum += ea[(long long)payS[q] * 8 + f];
    outEA[g * 8 + f] = valid ? sum : 0.0f;
    if (f == 0) {
      outEI[g] = valid ? (float)cs : -1.0f;
      outEI[(long long)e + g] = valid ? (float)cd : -1.0f;
    }
  } else {
    outEA[g * 8 + f] = 0.0f;
    if (f == 0) { outEI[g] = -1.0f; outEI[(long long)e + g] = -1.0f; }
  }
}

// ---------------- host side ----------------
static void scan_u32(const unsigned* in, unsigned* out, unsigned* bsums,
                     long long nLen, int mode, hipStream_t st) {
  int nb = (int)((nLen + STILE - 1) / STILE);
  k_scan_blocks<<<nb, BLK, 0, st>>>(in, out, bsums, nLen, mode);
  k_scan_carry<<<1, BLK, 0, st>>>(bsums, nb);
  k_scan_add<<<nb, BLK, 0, st>>>(out, bsums, nLen);
}

extern "C" void kernel_launch(void* const* d_in, const int* in_sizes, int n_in,
                              void* d_out, int out_size, void* d_ws, size_t ws_size,
                              hipStream_t stream) {
  const float* x = (const float*)d_in[0];
  const float* pos = (const float*)d_in[1];
  const int* batch = (const int*)d_in[2];
  const int* ei = (const int*)d_in[3];
  const float* ea = (const float*)d_in[4];
  int n = in_sizes[2];
  int e = in_sizes[3] / 2;

  float* out = (float*)d_out;
  float* out_x = out;                                   // n*64
  float* out_pos = out + (size_t)n * 64;                // n*3
  float* out_b = out_pos + (size_t)n * 3;               // n
  float* out_ei = out_b + (size_t)n;                    // 2*e
  float* out_ea = out_ei + (size_t)2 * e;               // e*8

  uint8_t* w = (uint8_t*)d_ws;
  size_t off = 0;
  auto alloc = [&](size_t bytes) -> void* {
    void* p = w + off;
    off += (bytes + 255) & ~(size_t)255;
    return p;
  };
  unsigned* slots = (unsigned*)alloc(64 * 4);
  unsigned* counts = (unsigned*)alloc((size_t)CAPV * 4);
  unsigned* P = (unsigned*)alloc((size_t)CAPV * 4);
  unsigned* raw = (unsigned*)alloc((size_t)n * 4);
  unsigned* cluster = (unsigned*)alloc((size_t)n * 4);
  unsigned* cnt = (unsigned*)alloc((size_t)n * 4);
  int* bmax = (int*)alloc((size_t)n * 4);
  unsigned long long* psum = (unsigned long long*)alloc((size_t)n * 3 * 8);
  unsigned long long* keyA = (unsigned long long*)alloc((size_t)e * 8);
  unsigned long long* keyB = (unsigned long long*)alloc((size_t)e * 8);
  unsigned* payA = (unsigned*)alloc((size_t)e * 4);
  unsigned* payB = (unsigned*)alloc((size_t)e * 4);
  int nbR = (e + STILE - 1) / STILE;
  unsigned* hist = (unsigned*)alloc((size_t)256 * nbR * 4);
  unsigned* histS = (unsigned*)alloc((size_t)256 * nbR * 4);
  unsigned* headA = (unsigned*)alloc((size_t)e * 4);
  unsigned* gidEx = (unsigned*)alloc((size_t)e * 4);
  unsigned* gstart = (unsigned*)alloc(((size_t)e + 1) * 4);
  unsigned* bsums = (unsigned*)alloc((size_t)8192 * 4);

  // init
  k_init_slots<<<1, 1, 0, stream>>>(slots);
  k_fill_u32<<<2048, BLK, 0, stream>>>(counts, 0u, (long long)CAPV);
  k_fill_u32<<<256, BLK, 0, stream>>>(cnt, 0u, (long long)n);
  k_fill_u32<<<256, BLK, 0, stream>>>((unsigned*)bmax, 0x80000000u, (long long)n);
  k_fill_u32<<<512, BLK, 0, stream>>>((unsigned*)psum, 0u, (long long)n * 6);
  k_fill_u32<<<2048, BLK, 0, stream>>>((unsigned*)out_x, 0u, (long long)n * 64);

  // nodes
  k_minmax<<<512, BLK, 0, stream>>>(pos, n, slots);
  k_raw<<<(n + BLK - 1) / BLK, BLK, 0, stream>>>(pos, batch, n, slots, raw, counts);
  scan_u32(counts, P, bsums, (long long)CAPV, 1, stream);
  k_total_p<<<1, 1, 0, stream>>>(P, counts, slots);
  k_cluster<<<(n + BLK - 1) / BLK, BLK, 0, stream>>>(raw, P, cluster, pos, batch, cnt, bmax, psum, n);
  long long n64 = (long long)n * 64;
  k_xmax<<<(int)((n64 + BLK - 1) / BLK), BLK, 0, stream>>>(x, cluster, (unsigned*)out_x, n64);
  k_xdec<<<(int)((n64 + BLK - 1) / BLK), BLK, 0, stream>>>(out_x, slots, n64);
  k_node_out<<<(n + BLK - 1) / BLK, BLK, 0, stream>>>(slots, cnt, bmax, psum, out_pos, out_b, n);

  // edges: key build + 5-pass stable LSD radix (40 bits cover (cs<<20)|cd)
  k_edge_key<<<(e + BLK - 1) / BLK, BLK, 0, stream>>>(ei, cluster, e, n, keyA, payA);
  unsigned long long *kin = keyA, *kout = keyB;
  unsigned *pin = payA, *pout = payB;
  for (int pass = 0; pass < 5; pass++) {
    int shift = pass * 8;
    k_rhist<<<nbR, BLK, 0, stream>>>(kin, (long long)e, shift, hist, nbR);
    scan_u32(hist, histS, bsums, (long long)256 * nbR, 0, stream);
    k_rscatter<<<nbR, BLK, 0, stream>>>(kin, pin, kout, pout, (long long)e, shift, histS, nbR);
    unsigned long long* tk = kin; kin = kout; kout = tk;
    unsigned* tp = pin; pin = pout; pout = tp;
  }
  // kin/pin now hold sorted keys/payloads
  k_heads<<<(e + BLK - 1) / BLK, BLK, 0, stream>>>(kin, headA, e);
  scan_u32(headA, gidEx, bsums, (long long)e, 0, stream);
  k_ngroups<<<1, 1, 0, stream>>>(headA, gidEx, gstart, slots, e);
  k_gstart<<<(e + BLK - 1) / BLK, BLK, 0, stream>>>(headA, gidEx, gstart, e);
  long long e8 = (long long)e * 8;
  k_groups<<<(int)((e8 + BLK - 1) / BLK), BLK, 0, stream>>>(kin, pin, gstart, slots, ea, out_ei, out_ea, e, n);
}